// MessagePassingLayer2_87110526697696
// MI455X (gfx1250) — compile-verified
//
#include <hip/hip_runtime.h>

// Problem constants from the reference
#define VN 100000   // nodes
#define TT 8        // edge types
#define EE 400000   // edges per type
#define DD 128      // feature dim

typedef float v2f __attribute__((ext_vector_type(2)));
typedef float v8f __attribute__((ext_vector_type(8)));

__device__ __forceinline__ void atomAddF(float* p, float v) {
  // relaxed agent-scope fp32 atomic -> global_atomic_add_f32 (L2-resident RMW)
  __hip_atomic_fetch_add(p, v, __ATOMIC_RELAXED, __HIP_MEMORY_SCOPE_AGENT);
}

__global__ void zero_kernel(float4* __restrict__ p, int n4) {
  int i = blockIdx.x * blockDim.x + threadIdx.x;
  if (i < n4) p[i] = make_float4(0.f, 0.f, 0.f, 0.f);
}

// One-time: Wtr[t][n][k] = W[t*128 + k][n]  so B fetches are K-contiguous (b64)
__global__ void transpose_w_kernel(const float* __restrict__ W, float* __restrict__ Wtr) {
  int i = blockIdx.x * blockDim.x + threadIdx.x;
  if (i < TT * DD * DD) {
    int t   = i >> 14;        // / (128*128)
    int rem = i & 16383;
    int n   = rem >> 7;
    int k   = rem & 127;
    Wtr[i] = W[(size_t)(t * DD + k) * DD + n];
  }
}

// Degree histograms for one edge type
__global__ void degree_kernel(const int* __restrict__ src, const int* __restrict__ dst,
                              float* __restrict__ deg_src, float* __restrict__ deg_dst) {
  int e = blockIdx.x * blockDim.x + threadIdx.x;
  if (e < EE) {
    atomAddF(&deg_src[src[e]], 1.0f);
    atomAddF(&deg_dst[dst[e]], 1.0f);
  }
}

// One wave32 per edge; each lane moves 4 contiguous floats.
// Edge index is wave-uniform -> scalarize the index/degree loads (SMEM path).
__global__ void scatter_kernel(const int* __restrict__ src, const int* __restrict__ dst,
                               const float* __restrict__ deg_src,
                               const float* __restrict__ nv,
                               float* __restrict__ agg) {
  int lane = threadIdx.x & 31;
  int e = __builtin_amdgcn_readfirstlane((blockIdx.x * blockDim.x + threadIdx.x) >> 5);
  if (e >= EE) return;                       // wave-uniform
  int s = __builtin_amdgcn_readfirstlane(src[e]);
  int d = __builtin_amdgcn_readfirstlane(dst[e]);
  float scale = __frsqrt_rn(fmaxf(deg_src[s], 1.0f));
  const float4 v = *(const float4*)(nv + (size_t)s * DD + lane * 4);
  float* ap = agg + (size_t)d * DD + lane * 4;
  atomAddF(ap + 0, v.x * scale);
  atomAddF(ap + 1, v.y * scale);
  atomAddF(ap + 2, v.z * scale);
  atomAddF(ap + 3, v.w * scale);
}

// Per-type GEMM accumulate on the fp32 matrix pipe (V_WMMA_F32_16X16X4_F32):
//   out[16-row tile] += (agg * rsqrt(max(deg_dst,1))) @ Wt
// One wave owns a 16-row M tile and all 8 N tiles (64 accumulator VGPRs);
// A is loaded once per wave, B comes from the transposed W (b64 per fetch).
// For t==0 the accumulator is seeded with the bias instead of reading out.
__global__ void __launch_bounds__(256) gemm_kernel(const float* __restrict__ agg,
                                                   const float* __restrict__ deg_dst,
                                                   const float* __restrict__ Wtr_t, // [128(n) x 128(k)]
                                                   const float* __restrict__ bias,
                                                   int first,
                                                   float* __restrict__ out) {
  const int lane  = threadIdx.x & 31;
  const int wv    = threadIdx.x >> 5;
  const int gwave = blockIdx.x * 8 + wv;
  const int mtiles = VN / 16;            // 6250, exact
  if (gwave >= mtiles) return;           // wave-uniform exit; EXEC stays all-1s

  const int v0   = gwave * 16;
  const int rc   = lane & 15;            // A: row, B/C: column within tile
  const int kh   = (lane >> 4) * 2;      // K sub-offset per half-wave (A and B)
  const int crow = (lane >> 4) * 8;      // C: +8 row offset for upper half-wave

  const float invnd = __frsqrt_rn(fmaxf(deg_dst[v0 + rc], 1.0f));

  v8f c[8];
  if (first) {
#pragma unroll
    for (int nt = 0; nt < 8; ++nt) {
      const float bv = bias[nt * 16 + rc];
#pragma unroll
      for (int r = 0; r < 8; ++r) c[nt][r] = bv;
    }
  } else {
#pragma unroll
    for (int nt = 0; nt < 8; ++nt) {
#pragma unroll
      for (int r = 0; r < 8; ++r)
        c[nt][r] = out[(size_t)(v0 + r + crow) * DD + nt * 16 + rc];
    }
  }

  const float* arow = agg + (size_t)(v0 + rc) * DD;

  for (int k0 = 0; k0 < DD; k0 += 4) {
    // A 16x4 f32: lanes 0-15 -> K = k0,k0+1 ; lanes 16-31 -> K = k0+2,k0+3
    const float2 av = *(const float2*)(arow + k0 + kh);
    v2f a;
    a.x = av.x * invnd;
    a.y = av.y * invnd;
#pragma unroll
    for (int nt = 0; nt < 8; ++nt) {
      // B 4x16 f32: N across lanes; K-contiguous in transposed W -> single b64
      const float2 wv2 = *(const float2*)(Wtr_t + (size_t)(nt * 16 + rc) * DD + k0 + kh);
      v2f bb;
      bb.x = wv2.x;
      bb.y = wv2.y;
      c[nt] = __builtin_amdgcn_wmma_f32_16x16x4_f32(
          /*neg_a=*/false, a, /*neg_b=*/false, bb,
          /*c_mod=*/(short)0, c[nt], /*reuse_a=*/false, /*reuse_b=*/false);
    }
  }

#pragma unroll
  for (int nt = 0; nt < 8; ++nt) {
#pragma unroll
    for (int r = 0; r < 8; ++r)
      out[(size_t)(v0 + r + crow) * DD + nt * 16 + rc] = c[nt][r];
  }
}

extern "C" void kernel_launch(void* const* d_in, const int* in_sizes, int n_in,
                              void* d_out, int out_size, void* d_ws, size_t ws_size,
                              hipStream_t stream) {
  const float* node_values = (const float*)d_in[0];  // [V, D]
  const int*   edges       = (const int*)d_in[1];    // [T, 2, E]
  const float* W           = (const float*)d_in[2];  // [T*D, D]
  const float* b           = (const float*)d_in[3];  // [D]
  float* out = (float*)d_out;                        // [V, D]

  // Workspace: deg_src[V] | deg_dst[V] | agg[V*D] | Wtr[T*D*D]   (~52.5 MB)
  float* deg_src = (float*)d_ws;
  float* deg_dst = deg_src + VN;
  float* agg     = deg_dst + VN;
  float* Wtr     = agg + (size_t)VN * DD;

  transpose_w_kernel<<<(TT * DD * DD + 255) / 256, 256, 0, stream>>>(W, Wtr);

  const int zero_n4 = (2 * VN + VN * DD) / 4;  // deg_src|deg_dst|agg are contiguous
  for (int t = 0; t < TT; ++t) {
    const int* src = edges + (size_t)t * 2 * EE;
    const int* dst = src + EE;

    zero_kernel<<<(zero_n4 + 255) / 256, 256, 0, stream>>>((float4*)d_ws, zero_n4);
    degree_kernel<<<(EE + 255) / 256, 256, 0, stream>>>(src, dst, deg_src, deg_dst);
    scatter_kernel<<<(EE * 32 + 255) / 256, 256, 0, stream>>>(src, dst, deg_src,
                                                              node_values, agg);
    gemm_kernel<<<(VN / 16 + 7) / 8, 256, 0, stream>>>(agg, deg_dst,
                                                       Wtr + (size_t)t * DD * DD,
                                                       b, (t == 0) ? 1 : 0, out);
  }
}